// GCN_Segmenter_35631048688033
// MI455X (gfx1250) — compile-verified
//
#include <hip/hip_runtime.h>

typedef __attribute__((ext_vector_type(8))) float v8f;
typedef __attribute__((ext_vector_type(2))) float v2f;

#define LN_EPS 1e-5f

// ---------------------------------------------------------------- utilities
__global__ void zero_kernel(float* __restrict__ p, long n) {
  long i = blockIdx.x * (long)blockDim.x + threadIdx.x;
  if (i < n) p[i] = 0.0f;
}

__global__ void degree_kernel(const int* __restrict__ dst, float* __restrict__ deg, int E) {
  int e = blockIdx.x * blockDim.x + threadIdx.x;
  if (e < E) atomicAdd(&deg[dst[e]], 1.0f);
}

__global__ void dinv_kernel(const float* __restrict__ deg, float* __restrict__ dis,
                            float* __restrict__ dinv, int N) {
  int i = blockIdx.x * blockDim.x + threadIdx.x;
  if (i < N) {
    float d = deg[i] + 1.0f;       // +1 for self loop
    dis[i]  = rsqrtf(d);
    dinv[i] = 1.0f / d;
  }
}

// ------------------------------------------------- dense GEMM via f32 WMMA
// C[M x 32] = A[M x K] * B[K x 32].  REQUIRES M % 16 == 0 (N=100000=6250*16).
// One wave per 16-row tile; two 16x16 f32 accumulators cover the 32 output
// columns; K stepped by 4 with V_WMMA_F32_16X16X4_F32.
//
// Lane layouts per CDNA5 ISA 7.12.2:
//   A (16x4): lanes 0-15 -> M=lane, K={k0,k0+1}; lanes 16-31 -> M=lane-16,
//             K={k0+2,k0+3}   (contiguous pair -> one global_load_b64)
//   B (4x16): lanes 0-15 -> N=lane, K={k0,k0+1}; lanes 16-31 -> N=lane-16,
//             K={k0+2,k0+3}   (contiguous in LDS-transposed B -> ds_load_b64)
//   C/D (16x16): vgpr v: lanes 0-15 -> M=v, lanes 16-31 -> M=v+8, N=lane&15
template <int K>
__global__ void __launch_bounds__(256)
gemm_wmma_f32(const float* __restrict__ A, const float* __restrict__ B,
              float* __restrict__ C, int M) {
  constexpr int BS = K + 2;              // even stride: keeps 8B alignment,
  __shared__ float Blds[32 * BS];        // breaks the 64-bank conflict pattern

  // cooperative transposed stage: Blds[c][k] = B[k][c]
  for (int idx = threadIdx.x; idx < 32 * K; idx += 256) {
    const int k = idx >> 5;
    const int c = idx & 31;
    Blds[c * BS + k] = B[idx];
  }
  __syncthreads();

  const int wave = (blockIdx.x * blockDim.x + threadIdx.x) >> 5;
  const int lane = threadIdx.x & 31;
  const int row0 = wave * 16;
  if (row0 >= M) return;                 // wave-uniform exit, EXEC stays full

  const int r     = lane & 15;           // row (A) / col (B) within tile
  const int khalf = (lane >> 4) * 2;     // 0 or 2
  const float* __restrict__ arow = A + (long)(row0 + r) * K + khalf;
  const float* __restrict__ b0p  = &Blds[r * BS + khalf];
  const float* __restrict__ b1p  = &Blds[(r + 16) * BS + khalf];

  v8f c0 = {};
  v8f c1 = {};

#pragma unroll
  for (int k0 = 0; k0 < K; k0 += 4) {
    const v2f a  = *(const v2f*)(arow + k0);   // global_load_b64
    const v2f b0 = *(const v2f*)(b0p + k0);    // ds_load_b64
    const v2f b1 = *(const v2f*)(b1p + k0);    // ds_load_b64
    c0 = __builtin_amdgcn_wmma_f32_16x16x4_f32(false, a, false, b0, (short)0, c0, false, false);
    c1 = __builtin_amdgcn_wmma_f32_16x16x4_f32(false, a, false, b1, (short)0, c1, false, false);
  }

  const int half = lane >> 4;
  float* __restrict__ crow = C + (long)row0 * 32 + r;
#pragma unroll
  for (int v = 0; v < 8; ++v) {
    const int rr = v + half * 8;
    crow[(long)rr * 32]      = c0[v];
    crow[(long)rr * 32 + 16] = c1[v];
  }
}

// --------------------------------------------- edge scatter (atomic in L2)
// Wave per edge, lane per channel: one coalesced 128B gather + one coalesced
// 128B atomic group per edge.  agg/xw fit in the 192MB L2, so these atomics
// never touch HBM.
__global__ void __launch_bounds__(256)
scatter32_kernel(const float* __restrict__ xw, const int* __restrict__ src,
                 const int* __restrict__ dst, const float* __restrict__ dis,
                 float* __restrict__ agg, int E) {
  const int e = (blockIdx.x * blockDim.x + threadIdx.x) >> 5;
  const int f = threadIdx.x & 31;
  if (e >= E) return;
  const int s = src[e];
  const int d = dst[e];
  const float coeff = dis[s] * dis[d];
  atomicAdd(&agg[(long)d * 32 + f], xw[(long)s * 32 + f] * coeff);
}

// thread per edge for the 2-channel layer; vectorized float2 gather
__global__ void __launch_bounds__(256)
scatter2_kernel(const float* __restrict__ xw3, const int* __restrict__ src,
                const int* __restrict__ dst, const float* __restrict__ dis,
                float* __restrict__ agg3, int E) {
  const int e = blockIdx.x * blockDim.x + threadIdx.x;
  if (e >= E) return;
  const int s = src[e];
  const int d = dst[e];
  const float coeff = dis[s] * dis[d];
  const v2f v = *(const v2f*)(xw3 + (long)s * 2);
  atomicAdd(&agg3[(long)d * 2 + 0], v.x * coeff);
  atomicAdd(&agg3[(long)d * 2 + 1], v.y * coeff);
}

// ---------------------- fused self-loop + bias + LayerNorm + ReLU (32 ch)
// One wave32 per node, lane == channel; mean/var via __shfl_xor tree.
__device__ __forceinline__ float wave_sum(float v) {
#pragma unroll
  for (int o = 16; o; o >>= 1) v += __shfl_xor(v, o, 32);
  return v;
}

__global__ void __launch_bounds__(256)
ln_relu_combine(const float* __restrict__ agg, const float* __restrict__ xw,
                const float* __restrict__ dinv, const float* __restrict__ bias,
                const float* __restrict__ g, const float* __restrict__ be,
                float* __restrict__ out, int N) {
  const int node = (blockIdx.x * blockDim.x + threadIdx.x) >> 5;
  const int lane = threadIdx.x & 31;
  if (node >= N) return;
  const long base = (long)node * 32;

  const float h  = agg[base + lane] + xw[base + lane] * dinv[node] + bias[lane];
  const float mu = wave_sum(h) * (1.0f / 32.0f);
  const float d  = h - mu;
  const float var = wave_sum(d * d) * (1.0f / 32.0f);
  const float y = d * rsqrtf(var + LN_EPS) * g[lane] + be[lane];
  out[base + lane] = fmaxf(y, 0.0f);
}

// layer-2 variant fused with the tiny 32->2 GEMM: skips materializing h2
__global__ void __launch_bounds__(256)
ln_relu_gemm3(const float* __restrict__ agg, const float* __restrict__ xw,
              const float* __restrict__ dinv, const float* __restrict__ bias,
              const float* __restrict__ g, const float* __restrict__ be,
              const float* __restrict__ W3, float* __restrict__ xw3, int N) {
  const int node = (blockIdx.x * blockDim.x + threadIdx.x) >> 5;
  const int lane = threadIdx.x & 31;
  if (node >= N) return;
  const long base = (long)node * 32;

  const float h  = agg[base + lane] + xw[base + lane] * dinv[node] + bias[lane];
  const float mu = wave_sum(h) * (1.0f / 32.0f);
  const float d  = h - mu;
  const float var = wave_sum(d * d) * (1.0f / 32.0f);
  const float y = fmaxf(d * rsqrtf(var + LN_EPS) * g[lane] + be[lane], 0.0f);

  const float p0 = wave_sum(y * W3[lane * 2 + 0]);
  const float p1 = wave_sum(y * W3[lane * 2 + 1]);
  if (lane == 0) {
    xw3[(long)node * 2 + 0] = p0;
    xw3[(long)node * 2 + 1] = p1;
  }
}

__global__ void final_combine(const float* __restrict__ agg3, const float* __restrict__ xw3,
                              const float* __restrict__ dinv, const float* __restrict__ b3,
                              float* __restrict__ out, int N) {
  const int idx = blockIdx.x * blockDim.x + threadIdx.x;
  if (idx < 2 * N) {
    const int i = idx >> 1;
    const int c = idx & 1;
    out[idx] = agg3[idx] + xw3[idx] * dinv[i] + b3[c];
  }
}

// ---------------------------------------------------------------- launcher
extern "C" void kernel_launch(void* const* d_in, const int* in_sizes, int n_in,
                              void* d_out, int out_size, void* d_ws, size_t ws_size,
                              hipStream_t stream) {
  const float* x   = (const float*)d_in[0];
  const int*   ei  = (const int*)d_in[1];
  const float* W1  = (const float*)d_in[2];
  const float* b1  = (const float*)d_in[3];
  const float* g1  = (const float*)d_in[4];
  const float* be1 = (const float*)d_in[5];
  const float* W2  = (const float*)d_in[6];
  const float* b2  = (const float*)d_in[7];
  const float* g2  = (const float*)d_in[8];
  const float* be2 = (const float*)d_in[9];
  const float* W3  = (const float*)d_in[10];
  const float* b3  = (const float*)d_in[11];
  float* out = (float*)d_out;

  const int N = in_sizes[0] / 128;   // 100000 (multiple of 16)
  const int E = in_sizes[1] / 2;     // 3200000
  const int* src = ei;
  const int* dst = ei + E;

  // workspace (floats): deg[N] dis[N] dinv[N] xw[32N] agg[32N] h[32N]
  //                     xw3[2N] agg3[2N]  -> ~103N floats ~ 41 MB
  float* ws   = (float*)d_ws;
  float* deg  = ws;
  float* dis  = deg + N;
  float* dinv = dis + N;
  float* xw   = dinv + N;
  float* agg  = xw + (long)N * 32;
  float* h    = agg + (long)N * 32;
  float* xw3  = h + (long)N * 32;
  float* agg3 = xw3 + (long)N * 2;

  auto cdiv = [](long a, long b) { return (int)((a + b - 1) / b); };
  const int BT = 256;

  // degrees
  zero_kernel<<<cdiv(N, BT), BT, 0, stream>>>(deg, N);
  degree_kernel<<<cdiv(E, BT), BT, 0, stream>>>(dst, deg, E);
  dinv_kernel<<<cdiv(N, BT), BT, 0, stream>>>(deg, dis, dinv, N);

  const int gemm_blocks  = cdiv(cdiv(N, 16), 8);   // 8 waves / 256-thread block
  const int edge_waves   = cdiv((long)E * 32, BT); // wave per edge

  // ---- layer 1: xw = x@W1 ; scatter ; h = relu(LN(agg + xw*dinv + b1))
  gemm_wmma_f32<128><<<gemm_blocks, BT, 0, stream>>>(x, W1, xw, N);
  zero_kernel<<<cdiv((long)N * 32, BT), BT, 0, stream>>>(agg, (long)N * 32);
  scatter32_kernel<<<edge_waves, BT, 0, stream>>>(xw, src, dst, dis, agg, E);
  ln_relu_combine<<<cdiv((long)N * 32, BT), BT, 0, stream>>>(agg, xw, dinv, b1, g1, be1, h, N);

  // ---- layer 2 (+ fused 32->2 GEMM producing xw3 directly)
  gemm_wmma_f32<32><<<gemm_blocks, BT, 0, stream>>>(h, W2, xw, N);
  zero_kernel<<<cdiv((long)N * 32, BT), BT, 0, stream>>>(agg, (long)N * 32);
  scatter32_kernel<<<edge_waves, BT, 0, stream>>>(xw, src, dst, dis, agg, E);
  ln_relu_gemm3<<<cdiv((long)N * 32, BT), BT, 0, stream>>>(agg, xw, dinv, b2, g2, be2, W3, xw3, N);

  // ---- layer 3: logits = scatter(xw3) + xw3*dinv + b3
  zero_kernel<<<cdiv((long)N * 2, BT), BT, 0, stream>>>(agg3, (long)N * 2);
  scatter2_kernel<<<cdiv(E, BT), BT, 0, stream>>>(xw3, src, dst, dis, agg3, E);
  final_combine<<<cdiv((long)N * 2, BT), BT, 0, stream>>>(agg3, xw3, dinv, b3, out, N);
}